// DeOverlapNet_34205119546056
// MI455X (gfx1250) — compile-verified
//
#include <hip/hip_runtime.h>
#include <hip/hip_bf16.h>

typedef __attribute__((ext_vector_type(2))) float v2f;
typedef __attribute__((ext_vector_type(8))) float v8f;

// ---------------- constants ----------------
// B=8, C=2, F=256, S=64, NWIN=128, 2*NWIN=256, ncoefs=4160, padded=4288
#define XI(t) ((t) + ((t) >> 6))   // LDS swizzle: +1 float pad every 64

// ws layout (floats)
#define Y_ELEMS   33554432u        // 8*2*256*64*128 raw stage-1 output
#define P1_ELEMS  65536u           // 512 blocks * 64 s * {sum,sumsq}
#define C1_ELEMS  256u             // (c,s) * {a1,b1}
#define P2_ELEMS  2560u            // 5 buckets * 256 blocks * {sum,sumsq}
#define C2_ELEMS  20u              // (bucket,c) * {a2,b2}

// CDNA5 async global->LDS (ASYNCcnt-tracked); GVS addressing: saddr + vaddr
__device__ __forceinline__ void async_ld_b128(uint32_t lds_off, uint32_t voff,
                                              const void* base) {
  asm volatile("global_load_async_to_lds_b128 %0, %1, %2"
               :: "v"(lds_off), "v"(voff), "s"((uint64_t)(uintptr_t)base)
               : "memory");
}
__device__ __forceinline__ void wait_async0() {
  asm volatile("s_wait_asynccnt 0x0" ::: "memory");
}

// =====================================================================
// Stage 1: deoverlap GEMM  y[b,c,f,s,n] = sum_k x_pad[b,c,f,s*64+k]*Wdeo[n,k]
// One block = (b, c, 8 f-rows); one wave = one f-row (32 16x16 tiles).
// LDS: Wdeo[128x256] (row pad 257, async-staged), 8 normalized+padded
// x rows (swizzled), 64x{sum,sumsq} stats.
// =====================================================================
__global__ __launch_bounds__(256) void k_deoverlap(
    const float* __restrict__ x, const float* __restrict__ imean,
    const float* __restrict__ iscale, const float* __restrict__ Wdeo,
    float* __restrict__ y, float* __restrict__ part1)
{
  extern __shared__ float sm[];
  float* sm_w = sm;                 // 128*257 = 32896
  float* sm_x = sm + 128 * 257;     // 8*4355  = 34840
  float* sm_s = sm_x + 8 * 4355;    // 128
  const int tid = threadIdx.x;
  const int g = blockIdx.x;               // ((b*2+c)*32 + fc)
  const int fc = g & 31, c = (g >> 5) & 1, b = g >> 6;
  const int f0 = fc * 8;

  // stage Wdeo into LDS (row-padded) via async DMA, one float4 per lane/iter
  {
    const uint32_t smw_base = (uint32_t)(uintptr_t)sm_w;
    for (int i = tid; i < 8192; i += 256) {       // i indexes float4
      const int n = i >> 6;                       // row (256 floats = 64 f4)
      const int k = (i & 63) * 4;
      async_ld_b128(smw_base + (uint32_t)(n * 257 + k) * 4u,
                    (uint32_t)i * 16u, Wdeo);
    }
  }
  // stage 8 x rows: input-normalize + HOP zero pad both ends + swizzle
  for (int fi = 0; fi < 8; ++fi) {
    const int f = f0 + fi;
    const float mu = imean[f], sc = iscale[f];
    const float* xr = x + (size_t)((b * 2 + c) * 256 + f) * 4160;
    float* dst = sm_x + fi * 4355;
    for (int t = tid; t < 4288; t += 256) {
      float v = 0.f;
      if (t >= 64 && t < 4224) v = (xr[t - 64] + mu) * sc;
      dst[XI(t)] = v;
    }
  }
  for (int i = tid; i < 128; i += 256) sm_s[i] = 0.f;
  wait_async0();
  __syncthreads();

  const int wave = tid >> 5, lane = tid & 31;
  const int row = lane & 15, hi = lane >> 4, kk = hi * 2;
  const int fi = wave;                       // wave <-> f-row
  const float* xr = sm_x + fi * 4355;
  const int f = f0 + fi;
  float* yr = y + (size_t)((b * 2 + c) * 256 + f) * 64 * 128;

  for (int st = 0; st < 4; ++st) {
    const int s0 = st * 16;
    const int arow = (s0 + row) * 64;        // overlap stride = HOP
    const float* ab = xr + arow + (arow >> 6);  // swizzled row base (linear per 64-blk)
    float ssum[8], ssq[8];
    #pragma unroll
    for (int r = 0; r < 8; ++r) { ssum[r] = 0.f; ssq[r] = 0.f; }

    #pragma unroll 1
    for (int nt = 0; nt < 8; ++nt) {
      const int n0 = nt * 16;
      const float* wr = sm_w + (n0 + row) * 257;
      v8f acc = {};
      #pragma unroll
      for (int kb = 0; kb < 4; ++kb) {       // 4 x 64-K blocks
        const float* a2 = ab + kb * 65;      // +65: 64 data + 1 pad
        const float* w2 = wr + kb * 64;
        #pragma unroll
        for (int kq = 0; kq < 64; kq += 4) { // 16 x v_wmma_f32_16x16x4_f32
          v2f a, bb;
          a.x  = a2[kq + kk];  a.y  = a2[kq + kk + 1];
          bb.x = w2[kq + kk];  bb.y = w2[kq + kk + 1];
          acc = __builtin_amdgcn_wmma_f32_16x16x4_f32(
              false, a, false, bb, (short)0, acc, false, false);
        }
      }
      #pragma unroll
      for (int r = 0; r < 8; ++r) {
        const float v = acc[r];
        yr[(size_t)(s0 + r + hi * 8) * 128 + n0 + row] = v;
        ssum[r] += v;
        ssq[r] += v * v;
      }
    }
    // one cross-lane (n) reduction per s-group instead of per tile
    #pragma unroll
    for (int r = 0; r < 8; ++r) {
      float sv = ssum[r], sq = ssq[r];
      #pragma unroll
      for (int m = 1; m < 16; m <<= 1) {
        sv += __shfl_xor(sv, m, 32);
        sq += __shfl_xor(sq, m, 32);
      }
      if (row == 0) {
        const int s = s0 + r + hi * 8;
        atomicAdd(&sm_s[2 * s], sv);
        atomicAdd(&sm_s[2 * s + 1], sq);
      }
    }
  }
  __syncthreads();
  for (int i = tid; i < 128; i += 256) part1[(size_t)g * 128 + i] = sm_s[i];
}

// ============ finalize BN1: per (c,s) affine coefs ============
__global__ void k_finalize1(const float* __restrict__ part1,
                            const float* __restrict__ gamma,
                            const float* __restrict__ beta,
                            float* __restrict__ C1)
{
  const int tid = threadIdx.x;
  if (tid >= 128) return;
  const int c = tid >> 6, s = tid & 63;
  float sum = 0.f, sq = 0.f;
  for (int b = 0; b < 8; ++b)
    for (int fcb = 0; fcb < 32; ++fcb) {
      const int g = (b * 2 + c) * 32 + fcb;
      sum += part1[(size_t)g * 128 + 2 * s];
      sq  += part1[(size_t)g * 128 + 2 * s + 1];
    }
  const float inv = 1.f / 262144.f;            // B*F*NWIN
  const float mu = sum * inv;
  const float var = sq * inv - mu * mu;
  const float a1 = gamma[c] * rsqrtf(var + 1e-5f);
  C1[tid * 2] = a1;
  C1[tid * 2 + 1] = beta[c] - mu * a1;
}

// =====================================================================
// Stage 2: per-bucket GEMM  z[s,m] = sum_t relu(a1*y[s,t]+b1) * Wdi[m,t]
// One block = (b, c, 4 f-rows); one wave = 2 (f,s-tile) groups.
// BN1+ReLU applied while staging y in LDS; Wdi async-staged.
// =====================================================================
__global__ __launch_bounds__(256) void k_deinterp(
    const float* __restrict__ y, const float* __restrict__ C1,
    const float* __restrict__ Wdi, float* __restrict__ zout,
    float* __restrict__ part2, int freqs, int mt, int p)
{
  extern __shared__ float sm[];
  float* sm_w = sm;                  // mt*129
  float* sm_y = sm + mt * 129;       // 4*64*129 = 33024
  float* sm_p = sm_y + 33024;        // 2
  const int tid = threadIdx.x;
  const int nfc = freqs >> 2;
  const int g = blockIdx.x;          // ((b*2+c)*nfc + fc)
  const int fc = g % nfc;
  const int c = (g / nfc) & 1;
  const int b = g / (nfc * 2);

  // stage Wdi via async DMA (rows of 128 floats = 32 float4, pad to 129)
  {
    const uint32_t smw_base = (uint32_t)(uintptr_t)sm_w;
    for (int i = tid; i < mt * 32; i += 256) {
      const int m = i >> 5;
      const int k = (i & 31) * 4;
      async_ld_b128(smw_base + (uint32_t)(m * 129 + k) * 4u,
                    (uint32_t)i * 16u, Wdi);
    }
  }
  for (int fi = 0; fi < 4; ++fi) {
    const int f = p + fc * 4 + fi;
    const float* yr = y + (size_t)((b * 2 + c) * 256 + f) * 64 * 128;
    float* dst = sm_y + fi * 8256;
    for (int i = tid; i < 8192; i += 256) {
      const int s = i >> 7, t = i & 127;
      const float a1 = C1[(c * 64 + s) * 2];
      const float b1 = C1[(c * 64 + s) * 2 + 1];
      const float v = fmaf(yr[i], a1, b1);
      dst[s * 129 + t] = v > 0.f ? v : 0.f;
    }
  }
  if (tid < 2) sm_p[tid] = 0.f;
  wait_async0();
  __syncthreads();

  const int wave = tid >> 5, lane = tid & 31;
  const int row = lane & 15, hi = lane >> 4, kk = hi * 2;
  const int mtiles = mt >> 4;
  float lsum = 0.f, lsq = 0.f;

  // 16 (fi,st) groups, 8 waves -> 2 groups per wave, mtiles m-tiles each
  for (int gi = 0; gi < 2; ++gi) {
    const int grp = wave * 2 + gi;
    const int fi = grp >> 2, st = grp & 3;
    const int s0 = st * 16;
    const float* ar = sm_y + fi * 8256 + (s0 + row) * 129;
    const int fl = fc * 4 + fi;
    float* zr = zout + (size_t)((b * 2 + c) * freqs + fl) * 64 * mt;
    #pragma unroll 1
    for (int ml = 0; ml < mtiles; ++ml) {
      const int m0 = ml * 16;
      const float* wr = sm_w + (m0 + row) * 129;
      v8f acc = {};
      #pragma unroll
      for (int k0 = 0; k0 < 128; k0 += 4) {  // 32 x v_wmma_f32_16x16x4_f32
        const int ka = k0 + kk;
        v2f a, bb;
        a.x = ar[ka];  a.y = ar[ka + 1];
        bb.x = wr[ka]; bb.y = wr[ka + 1];
        acc = __builtin_amdgcn_wmma_f32_16x16x4_f32(
            false, a, false, bb, (short)0, acc, false, false);
      }
      #pragma unroll
      for (int r = 0; r < 8; ++r) {
        const int s = s0 + r + hi * 8;
        const float v = acc[r];
        zr[(size_t)s * mt + m0 + row] = v;
        lsum += v;
        lsq += v * v;
      }
    }
  }
  #pragma unroll
  for (int m = 1; m < 32; m <<= 1) {
    lsum += __shfl_xor(lsum, m, 32);
    lsq  += __shfl_xor(lsq, m, 32);
  }
  if (lane == 0) { atomicAdd(&sm_p[0], lsum); atomicAdd(&sm_p[1], lsq); }
  __syncthreads();
  if (tid < 2) part2[(size_t)g * 2 + tid] = sm_p[tid];
}

// ============ finalize BN2: per (bucket,c) affine coefs ============
__global__ void k_finalize2(const float* __restrict__ part2,
    const float* g0, const float* g1, const float* g2, const float* g3, const float* g4,
    const float* b0, const float* b1, const float* b2, const float* b3, const float* b4,
    float* __restrict__ C2)
{
  const int tid = threadIdx.x;
  if (tid >= 10) return;
  const int bk = tid >> 1, c = tid & 1;
  const int FR[5] = {32, 32, 64, 64, 64};
  const int MT[5] = {16, 32, 48, 64, 128};
  const float* GS[5] = {g0, g1, g2, g3, g4};
  const float* BS[5] = {b0, b1, b2, b3, b4};
  const int freqs = FR[bk], mt = MT[bk];
  const int nfc = freqs >> 2;
  float sum = 0.f, sq = 0.f;
  for (int b = 0; b < 8; ++b)
    for (int fcb = 0; fcb < nfc; ++fcb) {
      const int g = (b * 2 + c) * nfc + fcb;
      sum += part2[(size_t)(bk * 256 + g) * 2];
      sq  += part2[(size_t)(bk * 256 + g) * 2 + 1];
    }
  const float inv = 1.f / ((float)(8 * 64) * (float)freqs * (float)mt);
  const float mu = sum * inv;
  const float var = sq * inv - mu * mu;
  const float a2 = GS[bk][c] * rsqrtf(var + 1e-5f);
  C2[tid * 2] = a2;
  C2[tid * 2 + 1] = BS[bk][c] - mu * a2;
}

// ============ apply BN2 + ReLU in place on d_out ============
__global__ void k_bnrelu2(float* __restrict__ z, const float* __restrict__ C2,
                          int bucket, int cs, int n)
{
  const int i = blockIdx.x * blockDim.x + threadIdx.x;
  if (i >= n) return;
  const int c = (i / cs) & 1;
  const float a2 = C2[(bucket * 2 + c) * 2];
  const float b2 = C2[(bucket * 2 + c) * 2 + 1];
  const float v = fmaf(z[i], a2, b2);
  z[i] = v > 0.f ? v : 0.f;
}

// =====================================================================
extern "C" void kernel_launch(void* const* d_in, const int* in_sizes, int n_in,
                              void* d_out, int out_size, void* d_ws, size_t ws_size,
                              hipStream_t stream) {
  (void)in_sizes; (void)n_in; (void)out_size; (void)ws_size;
  const float* x      = (const float*)d_in[0];
  const float* imean  = (const float*)d_in[1];
  const float* iscale = (const float*)d_in[2];
  const float* Wdeo   = (const float*)d_in[3];
  const float* g_deo  = (const float*)d_in[4];
  const float* b_deo  = (const float*)d_in[5];
  // d_in[6] = nb_slices (== 64, baked in)
  const float* Wdi[5]; const float* gdi[5]; const float* bdi[5];
  for (int i = 0; i < 5; ++i) {
    Wdi[i] = (const float*)d_in[7 + 3 * i];
    gdi[i] = (const float*)d_in[8 + 3 * i];
    bdi[i] = (const float*)d_in[9 + 3 * i];
  }

  float* ws    = (float*)d_ws;
  float* y     = ws;                       // Y_ELEMS
  float* part1 = y + Y_ELEMS;              // P1_ELEMS
  float* C1    = part1 + P1_ELEMS;         // C1_ELEMS
  float* part2 = C1 + C1_ELEMS;            // P2_ELEMS
  float* C2    = part2 + P2_ELEMS;         // C2_ELEMS
  float* out   = (float*)d_out;

  const size_t lds1 = (size_t)(128 * 257 + 8 * 4355 + 128) * sizeof(float);
  k_deoverlap<<<512, 256, lds1, stream>>>(x, imean, iscale, Wdeo, y, part1);
  k_finalize1<<<1, 128, 0, stream>>>(part1, g_deo, b_deo, C1);

  const int FR[5] = {32, 32, 64, 64, 64};
  const int MT[5] = {16, 32, 48, 64, 128};
  int p = 0;
  size_t off = 0;
  for (int bk = 0; bk < 5; ++bk) {
    const int freqs = FR[bk], mt = MT[bk];
    const int grid = 8 * 2 * (freqs >> 2);
    const size_t lds3 = (size_t)(mt * 129 + 33024 + 2) * sizeof(float);
    k_deinterp<<<grid, 256, lds3, stream>>>(y, C1, Wdi[bk], out + off,
                                            part2 + (size_t)bk * 512,
                                            freqs, mt, p);
    p += freqs;
    off += (size_t)8 * 2 * freqs * 64 * mt;
  }

  k_finalize2<<<1, 32, 0, stream>>>(part2,
      gdi[0], gdi[1], gdi[2], gdi[3], gdi[4],
      bdi[0], bdi[1], bdi[2], bdi[3], bdi[4], C2);

  off = 0;
  for (int bk = 0; bk < 5; ++bk) {
    const int freqs = FR[bk], mt = MT[bk];
    const int cs = freqs * 64 * mt;
    const int n = 8 * 2 * cs;
    k_bnrelu2<<<(n + 255) / 256, 256, 0, stream>>>(out + off, C2, bk, cs, n);
    off += (size_t)n;
  }
}